// CISP_9517647528057
// MI455X (gfx1250) — compile-verified
//
#include <hip/hip_runtime.h>
#include <math.h>

// ---------------------------------------------------------------------------
// Fused SSM step for MI455X (gfx1250, wave32):
//   per token: A = MLP_A(c) - eps*I (32x32), Bm = MLP_B(c) (32x16),
//   expA = expm(A*dt) (scaling & squaring, Taylor-8 Horner, fp32 WMMA),
//   x = (A + ridge I)^{-1} (expA - I) Bm (uW u + ub)   (Gauss-Jordan),
//   h_next = expA h + x.
//
// All dense math runs on V_WMMA_F32_16X16X4_F32:
//  - MLPs: computed transposed, Out^T = W @ X^T, M = output rows (full 16),
//    N = 16 token columns (4 live), K = 64.  A-operand = weight rows loaded
//    as one 8B global_load_b64 each; B-operand = activations staged in LDS
//    in a packed pair layout so each operand is one aligned ds_load_b64
//    (bank-conflict-free: half 0 -> banks 0-31, half 1 -> banks 32-63).
//  - expm: per-wave 32x32 matmuls, 2x2 tiles x 8 K-steps, register
//    accumulators so T can be squared in place; LDm=34 keeps every A-operand
//    an aligned single ds_load_b64.
// VALU is reserved for the Gauss-Jordan solve / matvecs (co-executes with
// the matrix pipe).
// ---------------------------------------------------------------------------

typedef __attribute__((ext_vector_type(2))) float v2f;
typedef __attribute__((ext_vector_type(8))) float v8f;

#define Bsz   8
#define Nn    4096
#define BN    (Bsz * Nn)
#define Dm    32
#define Pd    16
#define Cd    64
#define Hd    64
#define LDm   34          // even padded ld: 8B-aligned rows, conflict-free
#define LDb   17
#define EPSc  0.01f
#define RIDGEc 1e-4f
#define TOKS_PER_BLOCK 4

struct __align__(16) TokLds {
  float M[Dm * LDm];   // raw A -> A*dt*2^-s -> Gauss-Jordan system (rhs col 32)
  float T[Dm * LDm];   // Horner / squaring workspace -> expA
  float Bm[Dm * LDb];  // 32x16 B matrix
  float upv[Pd];       // u_proj(u)
  float tmpv[Dm];      // Bm @ up
  float hv[Dm];        // h
};

// Packed B-operand layout for 64-row x 16-col activation tiles:
// slot (kg, h, col) holds the pair {row 4kg+h, row 4kg+2+h} at `col`,
// i.e. exactly one fp32-WMMA B operand per 8B ds_load_b64.
__device__ __forceinline__ int packB(int r, int col) {
  return ((r >> 2) << 6) + ((r & 1) << 5) + (col << 1) + ((r >> 1) & 1);
}

// ---- WMMA fp32 16x16x4 operand loaders (ISA 7.12.2 fp32 layouts) ----------
// A (16x4): lane 0-15 hold M=lane%16, VGPR0 K={0|2}, VGPR1 K={1|3} by half.
__device__ __forceinline__ v2f ldA(const float* __restrict__ X, int row0, int k0, int lane) {
  const int m  = row0 + (lane & 15);
  const int kk = k0 + ((lane >> 4) << 1);
  return *(const v2f*)(X + m * LDm + kk);          // one ds_load_b64
}
// B (4x16): rows striped across lanes; VGPR0 rows {k0,k0+1}, VGPR1 {k0+2,k0+3}.
__device__ __forceinline__ v2f ldB(const float* __restrict__ Y, int k0, int col0, int lane) {
  const int n  = col0 + (lane & 15);
  const int kk = k0 + (lane >> 4);
  v2f b;
  b.x = Y[kk * LDm + n];
  b.y = Y[(kk + 2) * LDm + n];
  return b;
}

struct Acc4 { v8f t00, t01, t10, t11; };

// Full 32x32 * 32x32 product; accumulators in registers so dest may alias src.
__device__ __forceinline__ Acc4 matmul32(const float* __restrict__ X,
                                         const float* __restrict__ Y, int lane) {
  const v8f z = {0.f, 0.f, 0.f, 0.f, 0.f, 0.f, 0.f, 0.f};
  Acc4 r; r.t00 = z; r.t01 = z; r.t10 = z; r.t11 = z;
#pragma unroll
  for (int k8 = 0; k8 < 8; ++k8) {
    const int k0 = k8 * 4;
    v2f a0 = ldA(X, 0,  k0, lane);
    v2f a1 = ldA(X, 16, k0, lane);
    v2f b0 = ldB(Y, k0, 0,  lane);
    v2f b1 = ldB(Y, k0, 16, lane);
    r.t00 = __builtin_amdgcn_wmma_f32_16x16x4_f32(false, a0, false, b0, (short)0, r.t00, false, false);
    r.t01 = __builtin_amdgcn_wmma_f32_16x16x4_f32(false, a0, false, b1, (short)0, r.t01, false, false);
    r.t10 = __builtin_amdgcn_wmma_f32_16x16x4_f32(false, a1, false, b0, (short)0, r.t10, false, false);
    r.t11 = __builtin_amdgcn_wmma_f32_16x16x4_f32(false, a1, false, b1, (short)0, r.t11, false, false);
  }
  return r;
}

// C/D layout store: VGPR v holds row (v + 8*(lane/16)) of each 16x16 tile,
// col = lane%16.  Optionally scale and add identity.
__device__ __forceinline__ void store4(float* __restrict__ T, const Acc4& r, int lane,
                                       float scl, bool addI) {
  const int n  = lane & 15;
  const int hf = lane >> 4;
#pragma unroll
  for (int v = 0; v < 8; ++v) {
    const int m = v + 8 * hf;
    const float dg = (addI && m == n) ? 1.0f : 0.0f;
    T[m * LDm + n]             = r.t00[v] * scl + dg;
    T[m * LDm + n + 16]        = r.t01[v] * scl;
    T[(m + 16) * LDm + n]      = r.t10[v] * scl;
    T[(m + 16) * LDm + n + 16] = r.t11[v] * scl + dg;
  }
}

// ---- cooperative MLP tile: Out^T(16 rows x 16 tok) = W[row0..+15][0:64] @ Xb
// A-operand: weight rows from global, one 8B load per WMMA.
// B-operand: packed activation tile in LDS, one 8B ds load per WMMA.
__device__ __forceinline__ v8f mlp_tile(const float* __restrict__ W, int row0,
                                        const float* __restrict__ Xb, int lane) {
  const int m    = row0 + (lane & 15);
  const int half = lane >> 4;
  const int n2   = (lane & 15) << 1;
  v8f acc = {0.f, 0.f, 0.f, 0.f, 0.f, 0.f, 0.f, 0.f};
#pragma unroll
  for (int k8 = 0; k8 < 16; ++k8) {
    const float2 aw = *(const float2*)(W + m * 64 + k8 * 4 + 2 * half);
    v2f a; a.x = aw.x; a.y = aw.y;
    v2f b = *(const v2f*)(Xb + (k8 << 6) + (half << 5) + n2);
    acc = __builtin_amdgcn_wmma_f32_16x16x4_f32(false, a, false, b, (short)0, acc, false, false);
  }
  return acc;
}

__device__ __forceinline__ float silu(float x) { return x / (1.0f + __expf(-x)); }

__global__ __launch_bounds__(TOKS_PER_BLOCK * 32)
void ssm_expm_kernel(const float* __restrict__ h,  const float* __restrict__ u,
                     const float* __restrict__ c,  const float* __restrict__ dt,
                     const float* __restrict__ A_W1, const float* __restrict__ A_b1,
                     const float* __restrict__ A_W2, const float* __restrict__ A_b2,
                     const float* __restrict__ B_W1, const float* __restrict__ B_b1,
                     const float* __restrict__ B_W2, const float* __restrict__ B_b2,
                     const float* __restrict__ u_W,  const float* __restrict__ u_b,
                     float* __restrict__ out) {
  __shared__ TokLds lds[TOKS_PER_BLOCK];
  __shared__ float  sCB[Cd * 16];      // c^T,   packed B-operand layout (4KB)
  __shared__ float  sHidB[Hd * 16];    // hid^T, packed B-operand layout (4KB)

  const int wave = threadIdx.x >> 5;
  const int lane = threadIdx.x & 31;
  const int half = lane >> 4;
  const int n    = lane & 15;
  const int tok  = blockIdx.x * TOKS_PER_BLOCK + wave;   // grid is exact
  TokLds& L = lds[wave];

  const float dtb = dt[tok >> 12];          // N = 4096 tokens per batch row

  // ---- phase 0: zero-pad c^T, stage per-token vectors -------------------
  for (int idx = threadIdx.x; idx < Cd * 16; idx += TOKS_PER_BLOCK * 32)
    sCB[idx] = 0.0f;
  L.hv[lane] = h[tok * Dm + lane];
  {
    const int p = lane & 15;
    float acc = u_b[p];
    const float* uw = u_W + p * Pd;
    const float* ut = u + tok * Pd;
#pragma unroll
    for (int q = 0; q < Pd; ++q) acc += uw[q] * ut[q];
    if (lane < Pd) L.upv[lane] = acc;
  }
  __syncthreads();
  sCB[packB(lane,      wave)] = c[tok * Cd + lane];
  sCB[packB(32 + lane, wave)] = c[tok * Cd + 32 + lane];
  __syncthreads();

  // ---- MLP_A layer 1: hid^T = silu(A_W1 @ c^T + b1), M=64,N=16,K=64 -----
  {
    v8f t1 = mlp_tile(A_W1, wave * 16, sCB, lane);
#pragma unroll
    for (int v = 0; v < 8; ++v) {
      const int o = wave * 16 + v + 8 * half;
      sHidB[packB(o, n)] = silu(t1[v] + A_b1[o]);
    }
  }
  __syncthreads();

  // ---- MLP_A layer 2: A^T cols = A_W2 @ hid^T, M=1024,N=16,K=64 ---------
  for (int mt = 0; mt < 16; ++mt) {
    const int mtile = wave * 16 + mt;
    v8f ta = mlp_tile(A_W2, mtile * 16, sHidB, lane);
    if (n < TOKS_PER_BLOCK) {
#pragma unroll
      for (int v = 0; v < 8; ++v) {
        const int o = mtile * 16 + v + 8 * half;     // o = i*32 + j
        const int i = o >> 5, j = o & 31;
        lds[n].M[i * LDm + j] = ta[v] + A_b2[o] - ((i == j) ? EPSc : 0.0f);
      }
    }
  }
  __syncthreads();   // M complete; hid^T about to be recycled

  // ---- MLP_B layer 1 ----------------------------------------------------
  {
    v8f t1 = mlp_tile(B_W1, wave * 16, sCB, lane);
#pragma unroll
    for (int v = 0; v < 8; ++v) {
      const int o = wave * 16 + v + 8 * half;
      sHidB[packB(o, n)] = silu(t1[v] + B_b1[o]);
    }
  }
  __syncthreads();

  // ---- MLP_B layer 2: Bm^T cols = B_W2 @ hid^T, M=512,N=16,K=64 ---------
  for (int mt = 0; mt < 8; ++mt) {
    const int mtile = wave * 8 + mt;
    v8f tb = mlp_tile(B_W2, mtile * 16, sHidB, lane);
    if (n < TOKS_PER_BLOCK) {
#pragma unroll
      for (int v = 0; v < 8; ++v) {
        const int o = mtile * 16 + v + 8 * half;     // o = i*16 + p
        lds[n].Bm[(o >> 4) * LDb + (o & 15)] = tb[v] + B_b2[o];
      }
    }
  }
  __syncthreads();   // everything staged; waves now independent

  // ---- per-wave: pull own column of raw A, 1-norm, scale ----------------
  float Acol[Dm];                 // A(:,lane) incl. -eps*I, in registers
  float colsum = 0.0f;
#pragma unroll
  for (int i = 0; i < Dm; ++i) {
    Acol[i] = L.M[i * LDm + lane];
    colsum += fabsf(Acol[i]) * dtb;
  }
  float nrm = colsum;
#pragma unroll
  for (int off = 16; off > 0; off >>= 1) nrm = fmaxf(nrm, __shfl_xor(nrm, off, 32));
  int s = 0;
  if (nrm > 0.25f) s = (int)ceilf(log2f(nrm * 4.0f));   // ||M||_1 <= 0.25
  if (s > 24) s = 24;
  const float scl = dtb * exp2f(-(float)s);
#pragma unroll
  for (int i = 0; i < Dm; ++i) L.M[i * LDm + lane] = Acol[i] * scl;

  // ---- expm(M*2^s): Taylor-8 Horner, then s squarings (fp32 WMMA) -------
#pragma unroll
  for (int i = 0; i < Dm; ++i)
    L.T[i * LDm + lane] = L.M[i * LDm + lane] * 0.125f + ((i == lane) ? 1.0f : 0.0f);
#pragma unroll
  for (int j = 7; j >= 1; --j) {            // T = I + (M @ T)/j
    Acc4 p = matmul32(L.M, L.T, lane);
    store4(L.T, p, lane, 1.0f / (float)j, true);
  }
  for (int q = 0; q < s; ++q) {             // T = T @ T  (s wave-uniform)
    Acc4 p = matmul32(L.T, L.T, lane);
    store4(L.T, p, lane, 1.0f, false);
  }
  // T = expm(A*dt)

  // ---- rhs = (expA - I) @ (Bm @ up) -------------------------------------
  {
    float acc = 0.0f;
#pragma unroll
    for (int p = 0; p < Pd; ++p) acc += L.Bm[lane * LDb + p] * L.upv[p];
    L.tmpv[lane] = acc;
  }
#pragma unroll
  for (int i = 0; i < Dm; ++i)              // G = A + ridge*I (from registers)
    L.M[i * LDm + lane] = Acol[i] + ((i == lane) ? RIDGEc : 0.0f);
  {
    float acc = 0.0f;
#pragma unroll
    for (int j = 0; j < Dm; ++j) {
      const float e = L.T[lane * LDm + j] - ((lane == j) ? 1.0f : 0.0f);
      acc += e * L.tmpv[j];
    }
    L.M[lane * LDm + 32] = acc;             // rhs lives in the pad column
  }

  // ---- Gauss-Jordan solve (lane = row; ridge-regularized) ---------------
  for (int k = 0; k < Dm; ++k) {
    const float pinv = 1.0f / L.M[k * LDm + k];
    const float fac  = (lane == k) ? 0.0f : L.M[lane * LDm + k] * pinv;
    for (int j = k; j <= Dm; ++j)
      L.M[lane * LDm + j] -= fac * L.M[k * LDm + j];
  }
  const float x = L.M[lane * LDm + 32] / L.M[lane * LDm + lane];

  // ---- h_next = expA @ h + x --------------------------------------------
  float acc = x;
#pragma unroll
  for (int j = 0; j < Dm; ++j) acc += L.T[lane * LDm + j] * L.hv[j];
  out[tok * Dm + lane] = acc;
}

extern "C" void kernel_launch(void* const* d_in, const int* in_sizes, int n_in,
                              void* d_out, int out_size, void* d_ws, size_t ws_size,
                              hipStream_t stream) {
  (void)in_sizes; (void)n_in; (void)out_size; (void)d_ws; (void)ws_size;
  const float* h    = (const float*)d_in[0];
  const float* u    = (const float*)d_in[1];
  const float* c    = (const float*)d_in[2];
  const float* dt   = (const float*)d_in[3];
  const float* A_W1 = (const float*)d_in[4];
  const float* A_b1 = (const float*)d_in[5];
  const float* A_W2 = (const float*)d_in[6];
  const float* A_b2 = (const float*)d_in[7];
  const float* B_W1 = (const float*)d_in[8];
  const float* B_b1 = (const float*)d_in[9];
  const float* B_W2 = (const float*)d_in[10];
  const float* B_b2 = (const float*)d_in[11];
  const float* u_W  = (const float*)d_in[12];
  const float* u_b  = (const float*)d_in[13];

  dim3 grid(BN / TOKS_PER_BLOCK);
  dim3 block(TOKS_PER_BLOCK * 32);
  ssm_expm_kernel<<<grid, block, 0, stream>>>(h, u, c, dt,
                                              A_W1, A_b1, A_W2, A_b2,
                                              B_W1, B_b1, B_W2, B_b2,
                                              u_W, u_b, (float*)d_out);
}